// MSDeformSS_75359496176165
// MI455X (gfx1250) — compile-verified
//
#include <hip/hip_runtime.h>
#include <math.h>

// ---------------- problem constants (match reference) ----------------
#define D_MODEL 256
#define KHEADS  4
#define LVLS    4
#define PTS     4
#define D_INNER 64
#define N_STATE 16
#define DT_RANK 4
#define BB      2
#define NQ      4096
#define HWTOT   5440          // 64*64 + 32*32 + 16*16 + 8*8

#define SW_STRIDE (D_MODEL + 4)   // padded LDS row stride (floats) -> bank-conflict-free frags

typedef __attribute__((ext_vector_type(2))) float v2f;
typedef __attribute__((ext_vector_type(8))) float v8f;

// =====================================================================
// f32 WMMA GEMM:  C[M x N] = A[M x Kd] * W^T + bias   (W: [N x Kd])
// Block = 4 waves; block computes a 64(M) x 64(N) tile.
// W tile (64 x Kd) staged in LDS once per block and shared by all waves.
// A fragments double-buffered from global. V_WMMA_F32_16X16X4_F32.
// Requires: M % 64 == 0, N % 64 == 0, Kd == 256.
// =====================================================================
__global__ void __launch_bounds__(128)
msds_wmma_gemm_f32(const float* __restrict__ A, const float* __restrict__ W,
                   const float* __restrict__ bias, float* __restrict__ C,
                   int M, int N, int Kd)
{
    __shared__ float sW[64 * SW_STRIDE];   // 64 rows x (Kd + pad)

    const int wave  = threadIdx.x >> 5;
    const int lane  = threadIdx.x & 31;
    const int mTile = (blockIdx.x * 4 + wave) * 16;
    const int nTile = blockIdx.y * 64;

    // ---- cooperative stage of W[nTile .. nTile+63][0..Kd) into LDS ----
    {
        const int nvec = Kd >> 2;                    // float4 per row
        for (int i = threadIdx.x; i < 64 * nvec; i += 128) {
            const int row  = i / nvec;
            const int col4 = i - row * nvec;
            const float4 v = ((const float4*)(W + (size_t)(nTile + row) * Kd))[col4];
            ((float4*)&sW[row * SW_STRIDE])[col4] = v;
        }
    }
    __syncthreads();

    const int half = lane >> 4;   // 0: lanes 0-15, 1: lanes 16-31
    const int lid  = lane & 15;

    const float* Arow = A + (size_t)(mTile + lid) * Kd;
    const float* sB0  = &sW[(lid +  0) * SW_STRIDE];
    const float* sB1  = &sW[(lid + 16) * SW_STRIDE];
    const float* sB2  = &sW[(lid + 32) * SW_STRIDE];
    const float* sB3  = &sW[(lid + 48) * SW_STRIDE];

    v8f acc0 = {}, acc1 = {}, acc2 = {}, acc3 = {};

    // prologue: A fragment for k0 = 0
    v2f a_cur; a_cur.x = Arow[2 * half]; a_cur.y = Arow[2 * half + 1];

    #pragma unroll 2
    for (int k0 = 0; k0 < Kd; k0 += 4) {
        // prefetch next A fragment (clamped re-read on last iter, branchless)
        const int kn = (k0 + 4 < Kd) ? (k0 + 4) : k0;
        v2f a_nxt; a_nxt.x = Arow[kn + 2 * half]; a_nxt.y = Arow[kn + 2 * half + 1];

        // B fragments 4x16 from LDS: lanes0-15 -> K=k0,k0+1 ; lanes16-31 -> k0+2,k0+3
        const int ko = k0 + 2 * half;
        v2f b0; b0.x = sB0[ko]; b0.y = sB0[ko + 1];
        v2f b1; b1.x = sB1[ko]; b1.y = sB1[ko + 1];
        v2f b2; b2.x = sB2[ko]; b2.y = sB2[ko + 1];
        v2f b3; b3.x = sB3[ko]; b3.y = sB3[ko + 1];

        acc0 = __builtin_amdgcn_wmma_f32_16x16x4_f32(false, a_cur, false, b0, (short)0, acc0, false, false);
        acc1 = __builtin_amdgcn_wmma_f32_16x16x4_f32(false, a_cur, false, b1, (short)0, acc1, false, false);
        acc2 = __builtin_amdgcn_wmma_f32_16x16x4_f32(false, a_cur, false, b2, (short)0, acc2, false, false);
        acc3 = __builtin_amdgcn_wmma_f32_16x16x4_f32(false, a_cur, false, b3, (short)0, acc3, false, false);

        a_cur = a_nxt;
    }

    // C/D layout: VGPR r -> row r (lanes0-15) / row r+8 (lanes16-31), col = lid
    #pragma unroll
    for (int r = 0; r < 8; ++r) {
        const int row = mTile + r + 8 * half;
        float* crow = C + (size_t)row * N + nTile + lid;
        crow[ 0] = acc0[r] + bias[nTile + lid +  0];
        crow[16] = acc1[r] + bias[nTile + lid + 16];
        crow[32] = acc2[r] + bias[nTile + lid + 32];
        crow[48] = acc3[r] + bias[nTile + lid + 48];
    }
}

// =====================================================================
// Depthwise 3x3 SAME conv per level, layout (B, HWTOT, 256), ch = k*64+d.
// =====================================================================
__global__ void __launch_bounds__(256)
msds_dwconv(const float* __restrict__ val, const float* __restrict__ cw,
            const float* __restrict__ cb, float* __restrict__ out)
{
    const int LVL_W[4] = {64, 32, 16, 8};
    const int LVL_S[4] = {0, 4096, 5120, 5376};

    int idx = blockIdx.x * blockDim.x + threadIdx.x;
    const int total = BB * HWTOT * D_MODEL;
    if (idx >= total) return;

    const int c   = idx & (D_MODEL - 1);
    const int pix = (idx >> 8) % HWTOT;
    const int b   = idx / (HWTOT * D_MODEL);
    const int d   = c & (D_INNER - 1);

    int l = 0;
    if (pix >= 5376) l = 3; else if (pix >= 5120) l = 2; else if (pix >= 4096) l = 1;
    const int Wl = LVL_W[l], st0 = LVL_S[l];
    const int loc = pix - st0;
    const int y = loc / Wl, x = loc % Wl;

    const float* base = val + (size_t)b * HWTOT * D_MODEL;
    float sum = cb[d];
    #pragma unroll
    for (int ky = 0; ky < 3; ++ky) {
        const int yy = y + ky - 1;
        if (yy < 0 || yy >= Wl) continue;   // H == W per level
        #pragma unroll
        for (int kx = 0; kx < 3; ++kx) {
            const int xx = x + kx - 1;
            if (xx < 0 || xx >= Wl) continue;
            sum += cw[d * 9 + ky * 3 + kx] * base[(size_t)(st0 + yy * Wl + xx) * D_MODEL + c];
        }
    }
    out[idx] = sum;
}

// =====================================================================
// Fused bilinear sampling + selective scan + LayerNorm.
// 1 block = 1 (b,q); thread = (k = tid/64, d = tid%64).
// Only the final time step's y is needed (reference uses y[..., -1]).
// x_dbl for all (k,t,c) computed up front -> barrier-free scan loop.
// =====================================================================
__global__ void __launch_bounds__(256)
msds_sample_scan(const float* __restrict__ conv,   // (B, HWTOT, 256)
                 const float* __restrict__ offs,   // (B, NQ, 128)
                 const float* __restrict__ refp,   // (B, NQ, 4, 2)
                 const float* __restrict__ stbuf,  // (B*NQ, 256)
                 const float* __restrict__ xpw,    // (4, 36, 64)
                 const float* __restrict__ dtw,    // (4, 64, 4)
                 const float* __restrict__ dtb,    // (4, 64)
                 const float* __restrict__ Alog,   // (4, 64, 16)
                 const float* __restrict__ Dsp,    // (4, 64)
                 const float* __restrict__ ln_g, const float* __restrict__ ln_b,
                 float* __restrict__ ylast)        // (B*NQ, 256)
{
    __shared__ float s_xpw[KHEADS * 36 * D_INNER];   // 36 KB
    __shared__ float s_ss[KHEADS][17][D_INNER];      // 17 KB
    __shared__ float s_xd[KHEADS][17][36];           // 9.8 KB
    __shared__ float s_gx[KHEADS * LVLS * PTS];
    __shared__ float s_gy[KHEADS * LVLS * PTS];
    __shared__ float s_red[KHEADS][D_INNER];

    const int LVL_W[4] = {64, 32, 16, 8};
    const int LVL_S[4] = {0, 4096, 5120, 5376};

    const int bq  = blockIdx.x;          // 0 .. B*NQ-1
    const int b   = bq / NQ;
    const int q   = bq % NQ;
    const int tid = threadIdx.x;
    const int k   = tid >> 6;
    const int d   = tid & 63;

    for (int i = tid; i < KHEADS * 36 * D_INNER; i += 256) s_xpw[i] = xpw[i];

    if (tid < 64) {                      // 64 (k,l,p) combos
        const int l = (tid >> 2) & 3;
        const float Wl = (float)LVL_W[l];
        const float* op = offs + (size_t)(b * NQ + q) * 128 + tid * 2;
        const float* rp = refp + (size_t)(b * NQ + q) * (LVLS * 2) + l * 2;
        const float lx = rp[0] + op[0] / Wl;   // norm = (W,H), H==W per level
        const float ly = rp[1] + op[1] / Wl;
        s_gx[tid] = lx * Wl - 0.5f;            // 2*loc-1 -> pixel coords
        s_gy[tid] = ly * Wl - 0.5f;
    }
    __syncthreads();

    // ---------- bilinear sampling: ss[k][t][d], t = l*4 + p ----------
    const float* cbase = conv + (size_t)b * HWTOT * D_MODEL;
    #pragma unroll
    for (int t = 0; t < 16; ++t) {
        const int l = t >> 2;
        const int gi = k * 16 + t;       // (k*4+l)*4+p == k*16 + t
        const float gx = s_gx[gi], gy = s_gy[gi];
        const int   Wl = LVL_W[l], st0 = LVL_S[l];
        const float x0f = floorf(gx), y0f = floorf(gy);
        const int   x0 = (int)x0f, y0 = (int)y0f;
        const float wx = gx - x0f, wy = gy - y0f;
        float acc = 0.0f;
        #pragma unroll
        for (int ty = 0; ty < 2; ++ty) {
            const int iy = y0 + ty;
            if (iy < 0 || iy >= Wl) continue;
            const float wyv = ty ? wy : (1.0f - wy);
            #pragma unroll
            for (int tx = 0; tx < 2; ++tx) {
                const int ix = x0 + tx;
                if (ix < 0 || ix >= Wl) continue;
                const float w = (tx ? wx : (1.0f - wx)) * wyv;
                acc += w * cbase[(size_t)(st0 + iy * Wl + ix) * D_MODEL + k * D_INNER + d];
            }
        }
        s_ss[k][t][d] = acc;
    }
    s_ss[k][16][d] = stbuf[(size_t)bq * D_MODEL + k * D_INNER + d];
    __syncthreads();

    // -------- x_dbl for ALL (k,t,c) up front: 2448 dot-64 products ----
    for (int i = tid; i < KHEADS * 17 * 36; i += 256) {
        const int k2 = i / (17 * 36);
        const int r  = i - k2 * (17 * 36);
        const int t2 = r / 36;
        const int c2 = r - t2 * 36;
        const float* wrow = &s_xpw[(k2 * 36 + c2) * D_INNER];
        const float* urow = &s_ss[k2][t2][0];
        float s = 0.0f;
        #pragma unroll 8
        for (int dd = 0; dd < D_INNER; ++dd) s += wrow[dd] * urow[dd];
        s_xd[k2][t2][c2] = s;
    }
    __syncthreads();

    // per-thread constants
    float Arow[N_STATE];
    #pragma unroll
    for (int n = 0; n < N_STATE; ++n)
        Arow[n] = -expf(Alog[((k * D_INNER + d) * N_STATE) + n]);
    float dtr[DT_RANK];
    #pragma unroll
    for (int r = 0; r < DT_RANK; ++r)
        dtr[r] = dtw[(k * D_INNER + d) * DT_RANK + r];
    const float dtbv = dtb[k * D_INNER + d];
    const float Dsv  = Dsp[k * D_INNER + d];

    // ---------------- selective scan (barrier-free) -------------------
    float h[N_STATE];
    #pragma unroll
    for (int n = 0; n < N_STATE; ++n) h[n] = 0.0f;
    float yv = 0.0f;

    for (int t = 0; t < 17; ++t) {
        const float* xd = &s_xd[k][t][0];     // LDS broadcast across the k-group
        float dsum = dtbv;
        #pragma unroll
        for (int r = 0; r < DT_RANK; ++r) dsum += xd[r] * dtr[r];
        // stable softplus
        const float delta = (dsum > 0.0f) ? (dsum + log1pf(expf(-dsum)))
                                          : log1pf(expf(dsum));
        const float u  = s_ss[k][t][d];
        const float du = delta * u;
        #pragma unroll
        for (int n = 0; n < N_STATE; ++n)
            h[n] = expf(delta * Arow[n]) * h[n] + du * xd[4 + n];

        if (t == 16) {
            float s = 0.0f;
            #pragma unroll
            for (int n = 0; n < N_STATE; ++n) s += h[n] * xd[20 + n];
            yv = s + u * Dsv;
        }
    }

    // ------------------- LayerNorm over D per (k) --------------------
    s_red[k][d] = yv;
    __syncthreads();
    float s1 = 0.0f, s2 = 0.0f;
    #pragma unroll 8
    for (int i = 0; i < D_INNER; ++i) { const float v = s_red[k][i]; s1 += v; s2 += v * v; }
    const float mu  = s1 * (1.0f / 64.0f);
    const float var = s2 * (1.0f / 64.0f) - mu * mu;
    const float yn  = (yv - mu) * rsqrtf(var + 1e-5f) * ln_g[d] + ln_b[d];
    ylast[(size_t)bq * D_MODEL + k * D_INNER + d] = yn;
}

// =====================================================================
extern "C" void kernel_launch(void* const* d_in, const int* in_sizes, int n_in,
                              void* d_out, int out_size, void* d_ws, size_t ws_size,
                              hipStream_t stream)
{
    (void)in_sizes; (void)n_in; (void)out_size; (void)ws_size;

    const float* query   = (const float*)d_in[0];
    const float* refp    = (const float*)d_in[1];
    const float* inflat  = (const float*)d_in[2];
    // d_in[3], d_in[4]: spatial shapes / level starts (static, baked in)
    const float* W_value = (const float*)d_in[5];
    const float* b_value = (const float*)d_in[6];
    const float* W_offs  = (const float*)d_in[7];
    const float* b_offs  = (const float*)d_in[8];
    const float* W_query = (const float*)d_in[9];
    const float* b_query = (const float*)d_in[10];
    const float* W_out   = (const float*)d_in[11];
    const float* b_out   = (const float*)d_in[12];
    const float* conv_w  = (const float*)d_in[13];
    const float* conv_b  = (const float*)d_in[14];
    const float* xpw     = (const float*)d_in[15];
    const float* dtw     = (const float*)d_in[16];
    const float* dtb     = (const float*)d_in[17];
    const float* Alog    = (const float*)d_in[18];
    const float* Ds      = (const float*)d_in[19];
    const float* lng     = (const float*)d_in[20];
    const float* lnb     = (const float*)d_in[21];
    float* out = (float*)d_out;

    // workspace carve-out (floats)
    float* ws      = (float*)d_ws;
    float* v_val   = ws;                         // B*HW*256   = 2,785,280
    float* v_conv  = v_val  + (size_t)BB * HWTOT * D_MODEL;
    float* v_offs  = v_conv + (size_t)BB * HWTOT * D_MODEL;   // 8192*128
    float* v_st    = v_offs + (size_t)BB * NQ * 128;          // 8192*256
    float* v_y     = v_st   + (size_t)BB * NQ * D_MODEL;      // 8192*256

    // 1) value = input_flatten @ W_value^T + b   (M=10880, N=256, K=256)
    {
        dim3 g(10880 / 64, 256 / 64);
        msds_wmma_gemm_f32<<<g, 128, 0, stream>>>(inflat, W_value, b_value, v_val,
                                                  BB * HWTOT, D_MODEL, D_MODEL);
    }
    // 2) depthwise 3x3 conv per level
    {
        const int n = BB * HWTOT * D_MODEL;
        msds_dwconv<<<(n + 255) / 256, 256, 0, stream>>>(v_val, conv_w, conv_b, v_conv);
    }
    // 3) offsets = query @ W_offs^T + b          (M=8192, N=128, K=256)
    {
        dim3 g(BB * NQ / 64, 128 / 64);
        msds_wmma_gemm_f32<<<g, 128, 0, stream>>>(query, W_offs, b_offs, v_offs,
                                                  BB * NQ, 128, D_MODEL);
    }
    // 4) st = query @ W_query^T + b              (M=8192, N=256, K=256)
    {
        dim3 g(BB * NQ / 64, 256 / 64);
        msds_wmma_gemm_f32<<<g, 128, 0, stream>>>(query, W_query, b_query, v_st,
                                                  BB * NQ, D_MODEL, D_MODEL);
    }
    // 5) fused sampling + selective scan + LayerNorm
    msds_sample_scan<<<BB * NQ, 256, 0, stream>>>(v_conv, v_offs, refp, v_st,
                                                  xpw, dtw, dtb, Alog, Ds, lng, lnb, v_y);
    // 6) out = y_last @ W_out^T + b              (M=8192, N=256, K=256)
    {
        dim3 g(BB * NQ / 64, 256 / 64);
        msds_wmma_gemm_f32<<<g, 128, 0, stream>>>(v_y, W_out, b_out, out,
                                                  BB * NQ, D_MODEL, D_MODEL);
    }
}